// AttentionModel_34866544508947
// MI455X (gfx1250) — compile-verified
//
#include <hip/hip_runtime.h>
#include <math.h>

// Problem constants (match reference)
#define BB 256
#define NN 1000
#define DD 128
#define HH 8
#define DHH 16
#define NEGV (-1e9f)
#define NTILES 63          // ceil(1000/16)

typedef __attribute__((ext_vector_type(2))) float v2f;
typedef __attribute__((ext_vector_type(8))) float v8f;

__device__ __forceinline__ float red16_add(float t) {
  t += __shfl_xor(t, 1, 32);
  t += __shfl_xor(t, 2, 32);
  t += __shfl_xor(t, 4, 32);
  t += __shfl_xor(t, 8, 32);
  return t;  // sum within each 16-lane half
}

__global__ __launch_bounds__(256, 1)
void attn_step_kernel(const float* __restrict__ E,        // (B,N,D)
                      const float* __restrict__ Wnode,    // (D,3D)
                      const float* __restrict__ Wfixed,   // (D,D)
                      const float* __restrict__ Wstep,    // (2D,D)
                      const float* __restrict__ Wout,     // (D,D)
                      const int*   __restrict__ prev_idx, // (B,)
                      const int*   __restrict__ first_idx,// (B,)
                      const unsigned char* __restrict__ mask, // (B,N) bool
                      float* __restrict__ out)            // (B,N) log_p
{
  __shared__ float geL[DD];               // graph mean
  __shared__ float sfL[DD], spL[DD];      // first/prev node embeds
  __shared__ float qL[DD];                // query, pre-scaled by 1/sqrt(DH)
  __shared__ float hbuf[DD];              // merged attention heads
  __shared__ float gL[DD];                // glimpse
  __shared__ float uL[DD];                // W_node[:,2D:] @ glimpse, pre-scaled 1/sqrt(D)
  __shared__ float mM[8][HH], mS[8][HH], mAcc[8][HH][DHH];
  __shared__ float lg[NN];
  __shared__ float red[256];

  const int b    = blockIdx.x;
  const int tid  = threadIdx.x;
  const int wave = tid >> 5;
  const int lane = tid & 31;
  const int nlane = lane & 15;
  const int koff  = (lane >= 16) ? 2 : 0;   // A/B frag: upper half carries K=2,3

  const float* Eb   = E + (size_t)b * NN * DD;
  const unsigned char* mrow = mask + (size_t)b * NN;

  // ---------- phase 0a: zero mean accumulator ----------
  if (tid < DD) geL[tid] = 0.0f;
  __syncthreads();

  // ---------- phase 0b: graph mean (stream E once) ----------
  {
    float fs0 = 0.f, fs1 = 0.f, fs2 = 0.f, fs3 = 0.f;
    for (int n = wave; n < NN; n += 8) {
      const float* row = Eb + (size_t)n * DD + lane;
      fs0 += row[0]; fs1 += row[32]; fs2 += row[64]; fs3 += row[96];
    }
    atomicAdd(&geL[lane      ], fs0);
    atomicAdd(&geL[lane + 32 ], fs1);
    atomicAdd(&geL[lane + 64 ], fs2);
    atomicAdd(&geL[lane + 96 ], fs3);
  }
  __syncthreads();

  // ---------- phase 0c: step context gathers + finalize mean ----------
  if (tid < DD) {
    geL[tid] *= (1.0f / (float)NN);
    int fi = first_idx[b], pi = prev_idx[b];
    sfL[tid] = Eb[(size_t)fi * DD + tid];
    spL[tid] = Eb[(size_t)pi * DD + tid];
  }
  __syncthreads();

  // ---------- phase 0d: query = graph_mean@W_fixed + [first,prev]@W_step ----------
  if (tid < DD) {
    float acc = 0.f;
    for (int d = 0; d < DD; ++d) acc += geL[d] * Wfixed[d * DD + tid];
    for (int k = 0; k < DD; ++k) acc += sfL[k] * Wstep[k * DD + tid];
    for (int k = 0; k < DD; ++k) acc += spL[k] * Wstep[(DD + k) * DD + tid];
    qL[tid] = 0.25f * acc;   // fold 1/sqrt(DH=16)
  }
  __syncthreads();

  // ---------- pass A: fused K/V projection (f32 WMMA) + online-softmax attention ----------
  // Head-outer / tile-inner: per head the 64 B-fragments (K and V columns of W_node)
  // are loop-invariant and deliberately register-resident (128 VGPRs). Per tile the
  // inner loop is 32 aligned global b64 A-loads + 64 chained v_wmma_f32_16x16x4_f32.
#pragma unroll 1
  for (int h = 0; h < HH; ++h) {
    v2f bK[32], bV[32];
#pragma unroll
    for (int j = 0; j < 32; ++j) {
      const float* w0 = Wnode + (4 * j + koff) * 384 + h * 16 + nlane;
      bK[j].x = w0[0];         // W_node[4j+koff  , h*16+n]
      bK[j].y = w0[384];       // W_node[4j+koff+1, h*16+n]
      bV[j].x = w0[128];       // V columns: +128
      bV[j].y = w0[384 + 128];
    }
    const float qv = qL[h * 16 + nlane];
    float m_h = -INFINITY, s_h = 0.f, acc_h = 0.f;

#pragma unroll 1
    for (int tile = wave; tile < NTILES; tile += 8) {
      const int base = tile * 16;
      int node  = base + nlane;
      int nodec = node < NN ? node : (NN - 1);
      const float* arow = Eb + (size_t)nodec * DD + koff;

      // per-tile mask bits: bit r == masked(node base+r [lower half] / base+8+r [upper half])
      const int nodeh = base + ((lane >> 4) << 3);
      unsigned mb = 0;
#pragma unroll
      for (int r = 0; r < 8; ++r) {
        int nd = nodeh + r;
        mb |= ((unsigned)((nd < NN) ? (mrow[nd] != 0) : 1)) << r;
      }

      v8f ck = {}, cv = {};
#pragma unroll
      for (int j = 0; j < 32; ++j) {
        v2f a = *(const v2f*)(arow + 4 * j);   // 8B-aligned pair (K=4j+koff, +1)
        ck = __builtin_amdgcn_wmma_f32_16x16x4_f32(false, a, false, bK[j],
                                                   (short)0, ck, false, false);
        cv = __builtin_amdgcn_wmma_f32_16x16x4_f32(false, a, false, bV[j],
                                                   (short)0, cv, false, false);
      }

      // compat[node] = sum_dh Q[h,dh]*K[node,dh]  (C-frag: node r lanes0-15, node 8+r lanes16-31)
      float cr[8];
#pragma unroll
      for (int r = 0; r < 8; ++r) {
        float t = red16_add(ck[r] * qv);
        cr[r] = ((mb >> r) & 1u) ? NEGV : t;
      }
      // online softmax update
      float tmax = cr[0];
#pragma unroll
      for (int r = 1; r < 8; ++r) tmax = fmaxf(tmax, cr[r]);
      tmax = fmaxf(tmax, __shfl_xor(tmax, 16, 32));
      const float mnew  = fmaxf(m_h, tmax);
      const float scale = __expf(m_h - mnew);
      float p[8]; float psum = 0.f;
#pragma unroll
      for (int r = 0; r < 8; ++r) { p[r] = __expf(cr[r] - mnew); psum += p[r]; }
      psum += __shfl_xor(psum, 16, 32);
      s_h = s_h * scale + psum;
      m_h = mnew;
      // acc[dh] += sum_r p[node_r] * V[node_r, dh]; halves line up with p's halves
      float av = acc_h * scale;
#pragma unroll
      for (int r = 0; r < 8; ++r) av += p[r] * cv[r];
      acc_h = av;
    }

    // wave partial for this head -> LDS
    float accT = acc_h + __shfl_xor(acc_h, 16, 32);   // fold node halves; dh = lane (0..15)
    if (lane == 0) { mM[wave][h] = m_h; mS[wave][h] = s_h; }
    if (lane < 16) mAcc[wave][h][lane] = accT;
  }
  __syncthreads();

  // merge 8 waves per head, normalize -> heads
  if (tid < DD) {
    const int h = tid >> 4, dh = tid & 15;
    float M = -INFINITY;
    for (int w = 0; w < 8; ++w) M = fmaxf(M, mM[w][h]);
    float S = 0.f, hv = 0.f;
    for (int w = 0; w < 8; ++w) {
      float e = __expf(mM[w][h] - M);
      S  += mS[w][h] * e;
      hv += mAcc[w][h][dh] * e;
    }
    hbuf[h * 16 + dh] = hv / S;
  }
  __syncthreads();

  // glimpse = heads @ W_out
  if (tid < DD) {
    float g = 0.f;
    for (int k = 0; k < DD; ++k) g += hbuf[k] * Wout[k * DD + tid];
    gL[tid] = g;
  }
  __syncthreads();

  // u = W_node[:, 2D:3D] @ glimpse   (turns logits into a single matvec over E)
  if (tid < DD) {
    float u = 0.f;
    const float* wrow = Wnode + tid * 384 + 256;
    for (int d = 0; d < DD; ++d) u += wrow[d] * gL[d];
    uL[tid] = u * 0.08838834764831845f;   // fold 1/sqrt(D=128)
  }
  __syncthreads();

  // ---------- pass B: logits = E @ u, tanh clip, mask ----------
  for (int n = wave; n < NN; n += 8) {
    const float* row = Eb + (size_t)n * DD + lane;
    float d0 = row[0]  * uL[lane]      + row[32] * uL[lane + 32]
             + row[64] * uL[lane + 64] + row[96] * uL[lane + 96];
    d0 = red16_add(d0);
    d0 += __shfl_xor(d0, 16, 32);
    if (lane == 0) {
      float l = 10.0f * tanhf(d0);
      if (mrow[n]) l = NEGV;
      lg[n] = l;
    }
  }
  __syncthreads();

  // ---------- block log-softmax over N ----------
  float lm = -INFINITY;
  for (int n = tid; n < NN; n += 256) lm = fmaxf(lm, lg[n]);
  red[tid] = lm;
  __syncthreads();
  for (int s2 = 128; s2 > 0; s2 >>= 1) {
    if (tid < s2) red[tid] = fmaxf(red[tid], red[tid + s2]);
    __syncthreads();
  }
  const float Mx = red[0];
  __syncthreads();
  float ls = 0.f;
  for (int n = tid; n < NN; n += 256) ls += __expf(lg[n] - Mx);
  red[tid] = ls;
  __syncthreads();
  for (int s2 = 128; s2 > 0; s2 >>= 1) {
    if (tid < s2) red[tid] += red[tid + s2];
    __syncthreads();
  }
  const float lse = Mx + __logf(red[0]);
  for (int n = tid; n < NN; n += 256) out[(size_t)b * NN + n] = lg[n] - lse;
}

extern "C" void kernel_launch(void* const* d_in, const int* in_sizes, int n_in,
                              void* d_out, int out_size, void* d_ws, size_t ws_size,
                              hipStream_t stream) {
  (void)in_sizes; (void)n_in; (void)d_ws; (void)ws_size; (void)out_size;
  const float* E      = (const float*)d_in[0];
  const float* Wnode  = (const float*)d_in[1];
  const float* Wfixed = (const float*)d_in[2];
  const float* Wstep  = (const float*)d_in[3];
  const float* Wout   = (const float*)d_in[4];
  const int*   prev   = (const int*)d_in[5];
  const int*   first  = (const int*)d_in[6];
  const unsigned char* mask = (const unsigned char*)d_in[7];
  float* out = (float*)d_out;

  attn_step_kernel<<<BB, 256, 0, stream>>>(E, Wnode, Wfixed, Wstep, Wout,
                                           prev, first, mask, out);
}